// Estimation_88871463289511
// MI455X (gfx1250) — compile-verified
//
#include <hip/hip_runtime.h>

typedef __attribute__((ext_vector_type(2))) float v2f;
typedef __attribute__((ext_vector_type(8))) float v8f;

// ---------------------------------------------------------------------------
// Generic fp32 WMMA matmul: D[M x N] = A[M x 4*Ksteps] @ B[4*Ksteps x N]
// One 16x16 tile per wave, accumulated via V_WMMA_F32_16X16X4_F32.
// ---------------------------------------------------------------------------
__global__ void mm_wmma(const float* __restrict__ A, int lda,
                        const float* __restrict__ B, int ldb,
                        float* __restrict__ D, int ldd,
                        int Mtiles, int Ntiles, int Ksteps) {
  int wave = (int)((blockIdx.x * blockDim.x + threadIdx.x) >> 5);
  if (wave >= Mtiles * Ntiles) return;          // uniform per-wave: EXEC stays all-1s
  int lane = threadIdx.x & 31;
  int mt = wave / Ntiles, nt = wave % Ntiles;
  int half = lane >> 4, l15 = lane & 15;
  int arow = mt * 16 + l15;
  int bcol = nt * 16 + l15;
  v8f c = {};
  for (int ks = 0; ks < Ksteps; ++ks) {
    int k0 = ks * 4 + half * 2;
    v2f a, b;
    a.x = A[arow * lda + k0];
    a.y = A[arow * lda + k0 + 1];
    b.x = B[(k0    ) * ldb + bcol];
    b.y = B[(k0 + 1) * ldb + bcol];
    c = __builtin_amdgcn_wmma_f32_16x16x4_f32(false, a, false, b,
                                              (short)0, c, false, false);
  }
#pragma unroll
  for (int j = 0; j < 8; ++j)
    D[(mt * 16 + half * 8 + j) * ldd + bcol] = c[j];
}

// ---------------------------------------------------------------------------
// Prep: centers, invert A(20x20, RBF+1e-4 I), invert B(2x2),
// M_s[112x20] = rbf(ts,cen)@invA (rows>=100 zero), M_q[64x20],
// Vv[64x128] = mean-normalized v in (k,d)-major rows (rows>=40 zero).
// ---------------------------------------------------------------------------
__global__ void prep_kernel(const float* __restrict__ times,
                            const float* __restrict__ ts,
                            const float* __restrict__ bb,
                            const float* __restrict__ v,
                            const float* __restrict__ Bm,
                            const float* __restrict__ sig,
                            float* invA, float* invB, float* Ms, float* Mq,
                            float* Vv) {
  __shared__ float sA[20][20];
  __shared__ float cen[20];
  __shared__ float mean[40];
  const int tid = threadIdx.x;      // 128 threads
  const float sigma = sig[0];

  if (tid < 20) cen[tid] = 0.5f * (bb[tid] + bb[tid + 1]);
  __syncthreads();

  for (int e = tid; e < 400; e += 128) {
    int i = e / 20, j = e % 20;
    float dd = (cen[i] - cen[j]) / sigma;
    sA[i][j] = expf(-0.5f * dd * dd) + (i == j ? 1e-4f : 0.0f);
  }
  for (int r = tid; r < 40; r += 128) {
    int k = r >> 1, d = r & 1;
    float s = 0.f;
    for (int n = 0; n < 128; ++n) s += v[(k * 128 + n) * 2 + d];
    mean[r] = s * (1.0f / 128.0f);
  }
  __syncthreads();

  // In-place Gauss-Jordan inverse of sA (SPD, no pivoting needed)
  for (int p = 0; p < 20; ++p) {
    if (tid == p) {
      float pivinv = 1.0f / sA[p][p];
      for (int j = 0; j < 20; ++j) if (j != p) sA[p][j] *= pivinv;
      sA[p][p] = pivinv;
    }
    __syncthreads();
    if (tid < 20 && tid != p) {
      float f = sA[tid][p];
      for (int j = 0; j < 20; ++j) if (j != p) sA[tid][j] -= f * sA[p][j];
      sA[tid][p] = -f * sA[p][p];
    }
    __syncthreads();
  }

  for (int e = tid; e < 32 * 20; e += 128) {
    int i = e / 20, j = e % 20;
    invA[e] = (i < 20) ? sA[i][j] : 0.0f;
  }
  if (tid == 0) {
    float b00 = Bm[0], b01 = Bm[1], b10 = Bm[2], b11 = Bm[3];
    float inv = 1.0f / (b00 * b11 - b01 * b10);
    invB[0] = b11 * inv;  invB[1] = -b01 * inv;
    invB[2] = -b10 * inv; invB[3] = b00 * inv;
  }
  for (int e = tid; e < 112 * 20; e += 128) {
    int s = e / 20, k = e % 20;
    float acc = 0.f;
    if (s < 100) {
      float t = ts[s];
      for (int kp = 0; kp < 20; ++kp) {
        float dd = (t - cen[kp]) / sigma;
        acc += expf(-0.5f * dd * dd) * sA[kp][k];
      }
    }
    Ms[e] = acc;
  }
  for (int e = tid; e < 64 * 20; e += 128) {
    int s = e / 20, k = e % 20;
    float t = times[s];
    float acc = 0.f;
    for (int kp = 0; kp < 20; ++kp) {
      float dd = (t - cen[kp]) / sigma;
      acc += expf(-0.5f * dd * dd) * sA[kp][k];
    }
    Mq[e] = acc;
  }
  for (int e = tid; e < 64 * 128; e += 128) {
    int r = e >> 7, n = e & 127;
    float val = 0.f;
    if (r < 40) {
      int k = r >> 1, d = r & 1;
      val = v[(k * 128 + n) * 2 + d] - mean[r];
    }
    Vv[e] = val;
  }
}

// ---------------------------------------------------------------------------
// Invert 128x128 SPD C via in-place Gauss-Jordan in LDS (64 KB).
// ---------------------------------------------------------------------------
__global__ void invC_kernel(const float* __restrict__ Cm, float* __restrict__ iC) {
  __shared__ float sC[128][128];
  const int tid = threadIdx.x;      // 128 threads
  for (int e = tid; e < 128 * 128; e += 128) sC[e >> 7][e & 127] = Cm[e];
  __syncthreads();
  for (int p = 0; p < 128; ++p) {
    if (tid == p) {
      float pivinv = 1.0f / sC[p][p];
      for (int j = 0; j < 128; ++j) if (j != p) sC[p][j] *= pivinv;
      sC[p][p] = pivinv;
    }
    __syncthreads();
    if (tid != p) {
      float f = sC[tid][p];
      for (int j = 0; j < 128; ++j) if (j != p) sC[tid][j] -= f * sC[p][j];
      sC[tid][p] = -f * sC[p][p];
    }
    __syncthreads();
  }
  for (int e = tid; e < 128 * 128; e += 128) iC[e] = sC[e >> 7][e & 127];
}

// X2[(2k+d)][n] = sum_d' invB[d][d'] X1[(2k+d')][n]
__global__ void apply_invB(const float* __restrict__ X1,
                           const float* __restrict__ invB,
                           float* __restrict__ X2) {
  int e = blockIdx.x * blockDim.x + threadIdx.x;   // k*128+n, k<20
  if (e >= 20 * 128) return;
  int k = e >> 7, n = e & 127;
  float x0 = X1[(k * 2 + 0) * 128 + n];
  float x1 = X1[(k * 2 + 1) * 128 + n];
  X2[(k * 2 + 0) * 128 + n] = invB[0] * x0 + invB[1] * x1;
  X2[(k * 2 + 1) * 128 + n] = invB[2] * x0 + invB[3] * x1;
}

// G[k][d*128+n] = sum_d' B[d][d'] Y1[(2k+d')][n]
__global__ void apply_B_G(const float* __restrict__ Y1,
                          const float* __restrict__ Bm,
                          float* __restrict__ G) {
  int e = blockIdx.x * blockDim.x + threadIdx.x;   // k*128+n, k<20
  if (e >= 20 * 128) return;
  int k = e >> 7, n = e & 127;
  float y0 = Y1[(k * 2 + 0) * 128 + n];
  float y1 = Y1[(k * 2 + 1) * 128 + n];
  G[k * 256 + 0   + n] = Bm[0] * y0 + Bm[1] * y1;
  G[k * 256 + 128 + n] = Bm[2] * y0 + Bm[3] * y1;
}

// In-place cumulative sum of mv_s over s (rows 0..99), per column.
__global__ void cumsum_kernel(float* __restrict__ mvs) {
  int c = blockIdx.x * blockDim.x + threadIdx.x;
  if (c >= 256) return;
  float acc = 0.f;
  for (int s = 0; s < 100; ++s) {
    acc += mvs[s * 256 + c];
    mvs[s * 256 + c] = acc;
  }
}

// out[t,n,d] = dt * cumsum[idx(t)][d*128+n] + mv_q[t][d*128+n] * rem(t)
__global__ void final_kernel(const float* __restrict__ times,
                             const float* __restrict__ ts,
                             const float* __restrict__ mvcum,
                             const float* __restrict__ mvq,
                             float* __restrict__ out) {
  int gid = blockIdx.x * blockDim.x + threadIdx.x;
  if (gid >= 64 * 256) return;
  int t = gid >> 8, c = gid & 255;
  int d = c >> 7, n = c & 127;
  float q = times[t];
  float dt = ts[1] - ts[0];
  int idx = 0;
  for (int s = 1; s <= 98; ++s) idx += (ts[s] <= q) ? 1 : 0;   // searchsorted right
  float rem = q - ts[idx];
  out[t * 256 + n * 2 + d] = dt * mvcum[idx * 256 + c] + mvq[t * 256 + c] * rem;
}

// ---------------------------------------------------------------------------
extern "C" void kernel_launch(void* const* d_in, const int* in_sizes, int n_in,
                              void* d_out, int out_size, void* d_ws, size_t ws_size,
                              hipStream_t stream) {
  const float* times = (const float*)d_in[0];   // [64]
  const float* ts    = (const float*)d_in[1];   // [100]
  const float* bb    = (const float*)d_in[2];   // [21]
  const float* v     = (const float*)d_in[3];   // [20,128,2]
  const float* Bm    = (const float*)d_in[4];   // [2,2]
  const float* Cm    = (const float*)d_in[5];   // [128,128]
  const float* sig   = (const float*)d_in[6];   // [1]
  float* out = (float*)d_out;                   // [64,128,2]

  float* ws   = (float*)d_ws;
  float* invA = ws + 0;       // [32][20]   = 640
  float* invB = ws + 640;     // [4]
  float* Ms   = ws + 644;     // [112][20]  = 2240
  float* Mq   = ws + 2884;    // [64][20]   = 1280
  float* Vv   = ws + 4164;    // [64][128]  = 8192
  float* iC   = ws + 12356;   // [128][128] = 16384
  float* X1   = ws + 28740;   // [64][128]  = 8192
  float* X2   = ws + 36932;   // [64][128]  = 8192
  float* W    = ws + 45124;   // [64][128]  = 8192
  float* Y1   = ws + 53316;   // [64][128]  = 8192
  float* G    = ws + 61508;   // [32][256]  = 8192
  float* mvs  = ws + 69700;   // [112][256] = 28672
  float* mvq  = ws + 98372;   // [64][256]  = 16384

  prep_kernel<<<1, 128, 0, stream>>>(times, ts, bb, v, Bm, sig,
                                     invA, invB, Ms, Mq, Vv);
  invC_kernel<<<1, 128, 0, stream>>>(Cm, iC);

  // X1[48x128] = Vv @ invC   (24 tiles = 3 blocks x 8 waves)
  mm_wmma<<<3, 256, 0, stream>>>(Vv, 128, iC, 128, X1, 128, 3, 8, 32);
  apply_invB<<<10, 256, 0, stream>>>(X1, invB, X2);
  // W_d[32x128] = invA @ X2_d  for d=0,1  (16 tiles each = 2 blocks)
  mm_wmma<<<2, 256, 0, stream>>>(invA, 20, X2 + 0,   256, W + 0,   256, 2, 8, 5);
  mm_wmma<<<2, 256, 0, stream>>>(invA, 20, X2 + 128, 256, W + 128, 256, 2, 8, 5);
  // Y1[48x128] = W @ C
  mm_wmma<<<3, 256, 0, stream>>>(W, 128, Cm, 128, Y1, 128, 3, 8, 32);
  apply_B_G<<<10, 256, 0, stream>>>(Y1, Bm, G);
  // mv_s[112x256] = M_s @ G   (112 tiles = 14 blocks)
  mm_wmma<<<14, 256, 0, stream>>>(Ms, 20, G, 256, mvs, 256, 7, 16, 5);
  // mv_q[64x256]  = M_q @ G   (64 tiles = 8 blocks)
  mm_wmma<<<8, 256, 0, stream>>>(Mq, 20, G, 256, mvq, 256, 4, 16, 5);

  cumsum_kernel<<<1, 256, 0, stream>>>(mvs);
  final_kernel<<<64, 256, 0, stream>>>(times, ts, mvs, mvq, out);
}